// VoiceSpeaker_85572928405749
// MI455X (gfx1250) — compile-verified
//
#include <hip/hip_runtime.h>
#include <hip/hip_bf16.h>

typedef float v2f __attribute__((ext_vector_type(2)));
typedef float v8f __attribute__((ext_vector_type(8)));

// ---------------- constants ----------------
#define NFFT   1024
#define HOP    256
#define NT     16384            // frames used (stft frame 0 dropped)
#define TS     (HOP * NT)       // 4194304 signal samples
#define BINS   513
#define NS     1040             // complex-spectrum row stride (2*513 -> 65 tiles)
#define NC     528              // cepstrum row stride (513 -> 33 tiles)
#define KP_STFT 1024
#define KP_MP   576             // 513 padded to 9 chunks of 64
#define KP_IST  1088            // 1026 padded to 17 chunks of 64
#define OUTLEN ((NT - 1) * HOP) // 4194048
#define INV_SR (1.0f / 24000.0f)

static __device__ __forceinline__ float hannf(int n) {
  return 0.5f * (1.0f - cospif((float)n / 512.0f));
}

// ---------------- WMMA core (fp32, 16x16x4) ----------------
static __device__ __forceinline__ v8f wmma4(v2f a, v2f b, v8f c) {
#if __has_builtin(__builtin_amdgcn_wmma_f32_16x16x4_f32)
  return __builtin_amdgcn_wmma_f32_16x16x4_f32(false, a, false, b, (short)0, c,
                                               false, false);
#else
  c[0] += a.x * b.x + a.y * b.y; // compile fallback only
  return c;
#endif
}

// B matrices are stored K-pair-interleaved: element (k,n) lives at
//   (k>>1)*(2*NSTRIDE) + 2*n + (k&1)
// so one aligned b64 load yields the {K, K+1} fragment pair for a lane.
//
// Generic GEMM: OUT[M=16384][NSTRIDE] = A[M][K] * B[K][NSTRIDE]
// AMODE 0: A row-major  (A[gm*astride + gk], gk < aguard)
// AMODE 1: A transposed (A[gk*astride + gm], gk < aguard)  [bin-major envelopes]
// AMODE 2: A = reflect-padded STFT frames of signal A[TS]
// Each wave computes two adjacent 16x16 tiles (A fragment reused).
template <int AMODE, int NSTRIDE>
__global__ __launch_bounds__(256) void k_gemm(const float* __restrict__ A,
                                              const float* __restrict__ B,
                                              float* __restrict__ OUT, int kpad,
                                              int ntiles, int astride,
                                              int aguard, int accumulate) {
  __shared__ __align__(16) float As[16 * 68]; // stride 68: conflict-free
  const int tid  = threadIdx.x;
  const int lane = tid & 31;
  const int wave = tid >> 5;
  const int lm   = lane & 15;
  const int half = lane >> 4;
  const int Mbase = blockIdx.y << 4;
  const int t0 = (blockIdx.x * 8 + wave) * 2; // two adjacent N tiles per wave
  const bool act0 = (t0 < ntiles);
  const bool act1 = (t0 + 1 < ntiles);
  const int ncol0 = (t0 << 4) + lm;
  v8f acc0 = {0.f, 0.f, 0.f, 0.f, 0.f, 0.f, 0.f, 0.f};
  v8f acc1 = {0.f, 0.f, 0.f, 0.f, 0.f, 0.f, 0.f, 0.f};

  for (int kb = 0; kb < kpad; kb += 64) {
    // cooperative load of 16x64 A panel into LDS
    for (int e = tid; e < 1024; e += 256) {
      int m = e >> 6, k = e & 63;
      int gk = kb + k, gm = Mbase + m;
      float val;
      if (AMODE == 0) {
        val = (gk < aguard) ? A[(size_t)gm * astride + gk] : 0.0f;
      } else if (AMODE == 1) {
        val = (gk < aguard) ? A[(size_t)gk * astride + gm] : 0.0f;
      } else {
        int p = gm * HOP + gk - 256; // frame (t+1), center pad 512
        if (p < 0) p = -p;
        else if (p >= TS) p = 2 * TS - 2 - p;
        val = A[p];
      }
      As[m * 68 + k] = val;
    }
    __syncthreads();
    if (act0) {
      // base of the {K,K+1} pair rows for this chunk; kk advances by
      // kk*NSTRIDE floats -> compile-time immediate offsets.
      const float* Bp = B + (size_t)((kb >> 1) + half) * (2 * NSTRIDE) +
                        2 * ncol0;
      const float* Ap = &As[lm * 68 + 2 * half];
      if (kb + 64 < kpad)
        __builtin_prefetch(B + (size_t)((kb + 64) >> 1) * (2 * NSTRIDE) +
                               2 * ncol0, 0, 1);
      if (act1) {
#pragma unroll
        for (int kk = 0; kk < 64; kk += 4) {
          v2f a  = *(const v2f*)(Ap + kk); // K = kk + 2*half + {0,1}
          v2f b0 = *(const v2f*)(Bp + (size_t)kk * NSTRIDE);
          v2f b1 = *(const v2f*)(Bp + (size_t)kk * NSTRIDE + 32);
          acc0 = wmma4(a, b0, acc0);
          acc1 = wmma4(a, b1, acc1);
        }
      } else {
#pragma unroll
        for (int kk = 0; kk < 64; kk += 4) {
          v2f a  = *(const v2f*)(Ap + kk);
          v2f b0 = *(const v2f*)(Bp + (size_t)kk * NSTRIDE);
          acc0 = wmma4(a, b0, acc0);
        }
      }
    }
    __syncthreads();
  }
  if (act0) {
#pragma unroll
    for (int v = 0; v < 8; ++v) { // C/D layout: VGPR v -> M = v + 8*half
      size_t o = (size_t)(Mbase + v + 8 * half) * NSTRIDE + ncol0;
      if (accumulate) {
        OUT[o] += acc0[v];
        if (act1) OUT[o + 16] += acc1[v];
      } else {
        OUT[o] = acc0[v];
        if (act1) OUT[o + 16] = acc1[v];
      }
    }
  }
}

// ---------------- DFT / window matrix builders ----------------
// All builders write the K-pair-interleaved layout: storage offset idx
// decodes as pair = idx/(2*NSTRIDE); col = (idx%(2*NSTRIDE))>>1;
// krow = 2*pair + (idx&1).
__global__ void k_build_bfwd(float* B) { // logical [1024 time][NS]: fwd rDFT*w
  for (int idx = blockIdx.x * blockDim.x + threadIdx.x; idx < KP_STFT * NS;
       idx += gridDim.x * blockDim.x) {
    int pr = idx / (2 * NS), rem = idx - pr * (2 * NS);
    int c = rem >> 1;             // column 0..NS-1
    int n = 2 * pr + (rem & 1);   // time sample row
    float v = 0.0f;
    if (c < 2 * BINS) {
      int k = c >> 1;
      float x = (float)((k * n) & 1023) / 512.0f; // exact mod-2pi reduction
      float w = hannf(n);
      v = (c & 1) ? (-w * sinpif(x)) : (w * cospif(x));
    }
    B[idx] = v;
  }
}
__global__ void k_build_b1(float* B) { // logical [KP_MP bin][NC]: irfft + fold
  for (int idx = blockIdx.x * blockDim.x + threadIdx.x; idx < KP_MP * NC;
       idx += gridDim.x * blockDim.x) {
    int pr = idx / (2 * NC), rem = idx - pr * (2 * NC);
    int n = rem >> 1;             // cepstrum index column
    int k = 2 * pr + (rem & 1);   // bin row
    float v = 0.0f;
    if (k < BINS && n < BINS) {
      float ck = (k == 0 || k == 512) ? 1.0f : 2.0f;
      float fn = (n == 0 || n == 512) ? 1.0f : 2.0f;
      float x = (float)((k * n) & 1023) / 512.0f;
      v = ck * fn * (1.0f / 1024.0f) * cospif(x);
    }
    B[idx] = v;
  }
}
__global__ void k_build_b2(float* B) { // logical [KP_MP cep][NS]: rfft
  for (int idx = blockIdx.x * blockDim.x + threadIdx.x; idx < KP_MP * NS;
       idx += gridDim.x * blockDim.x) {
    int pr = idx / (2 * NS), rem = idx - pr * (2 * NS);
    int c = rem >> 1;
    int n = 2 * pr + (rem & 1);   // cepstrum index row
    float v = 0.0f;
    if (n < BINS && c < 2 * BINS) {
      int k = c >> 1;
      float x = (float)((k * n) & 1023) / 512.0f;
      v = (c & 1) ? -sinpif(x) : cospif(x);
    }
    B[idx] = v;
  }
}
__global__ void k_build_binv(float* B) { // logical [KP_IST re/im][1024]: irfft*w
  for (int idx = blockIdx.x * blockDim.x + threadIdx.x; idx < KP_IST * NFFT;
       idx += gridDim.x * blockDim.x) {
    int pr = idx / (2 * NFFT), rem = idx - pr * (2 * NFFT);
    int n = rem >> 1;             // time sample column
    int r = 2 * pr + (rem & 1);   // interleaved re/im row
    float v = 0.0f;
    if (r < 2 * BINS) {
      int k = r >> 1;
      float ck = ((k == 0 || k == 512) ? 1.0f : 2.0f) * (1.0f / 1024.0f);
      float x = (float)((k * n) & 1023) / 512.0f;
      v = hannf(n) * ck * ((r & 1) ? -sinpif(x) : cospif(x));
    }
    B[idx] = v;
  }
}

// ---------------- impulse train (hierarchical cumsum) ----------------
static __device__ __forceinline__ float f0up_val(const float* lf0, int i) {
  float pos = ((float)i + 0.5f) * (1.0f / 256.0f) - 0.5f;
  pos = fminf(fmaxf(pos, 0.0f), 16383.0f);
  int lo = (int)pos;
  int hi = lo + 1 > 16383 ? 16383 : lo + 1;
  float frac = pos - (float)lo;
  return expf(lf0[lo]) * (1.0f - frac) + expf(lf0[hi]) * frac;
}

__global__ __launch_bounds__(256) void k_f0a(const float* __restrict__ lf0,
                                             float* __restrict__ f0up,
                                             float* __restrict__ bsums) {
  __shared__ float red[256];
  int tid = threadIdx.x;
  int base = blockIdx.x * 4096 + tid * 16;
  float s = 0.f;
  for (int j = 0; j < 16; ++j) {
    float v = f0up_val(lf0, base + j);
    f0up[base + j] = v;
    s += v;
  }
  red[tid] = s;
  __syncthreads();
  for (int off = 128; off > 0; off >>= 1) {
    if (tid < off) red[tid] += red[tid + off];
    __syncthreads();
  }
  if (tid == 0) bsums[blockIdx.x] = red[0];
}

__global__ __launch_bounds__(1024) void k_scan(const float* __restrict__ bsums,
                                               float* __restrict__ boffs) {
  __shared__ float s[1024];
  int tid = threadIdx.x;
  float v = bsums[tid];
  s[tid] = v;
  __syncthreads();
  for (int off = 1; off < 1024; off <<= 1) {
    float t = (tid >= off) ? s[tid - off] : 0.f;
    __syncthreads();
    s[tid] += t;
    __syncthreads();
  }
  boffs[tid] = s[tid] - v; // exclusive
}

__global__ __launch_bounds__(256) void k_saw(const float* __restrict__ f0up,
                                             const float* __restrict__ boffs,
                                             float* __restrict__ saw) {
  __shared__ float ts[256];
  int tid = threadIdx.x;
  int base = blockIdx.x * 4096 + tid * 16;
  float loc[16];
  float s = 0.f;
  for (int j = 0; j < 16; ++j) { loc[j] = f0up[base + j]; s += loc[j]; }
  ts[tid] = s;
  __syncthreads();
  for (int off = 1; off < 256; off <<= 1) {
    float t = (tid >= off) ? ts[tid - off] : 0.f;
    __syncthreads();
    ts[tid] += t;
    __syncthreads();
  }
  float run = boffs[blockIdx.x] + ts[tid] - s;
  for (int j = 0; j < 16; ++j) {
    run += loc[j];
    float ph = run * INV_SR;
    saw[base + j] = ph - floorf(ph);
  }
}

__global__ void k_pulse(const float* __restrict__ saw,
                        const float* __restrict__ f0up,
                        float* __restrict__ pulse) {
  for (int i = blockIdx.x * blockDim.x + threadIdx.x; i < TS;
       i += gridDim.x * blockDim.x) {
    int nx = (i + 1 == TS) ? 0 : i + 1;
    pulse[i] = saw[i] - saw[nx] + f0up[i] * INV_SR;
  }
}

// ---------------- spectral pointwise ops ----------------
__global__ void k_envnoi(const float* __restrict__ env, float* __restrict__ S) {
  int n = NT * BINS;
  for (int idx = blockIdx.x * blockDim.x + threadIdx.x; idx < n;
       idx += gridDim.x * blockDim.x) {
    int k = idx >> 14, t = idx & (NT - 1); // env is [bin][frame]
    float s = expf(env[idx]);
    size_t o = (size_t)t * NS + 2 * k;
    S[o] *= s;
    S[o + 1] *= s;
  }
}

__global__ void k_minph(const float* __restrict__ Ms, float* __restrict__ S) {
  int n = NT * BINS;
  for (int idx = blockIdx.x * blockDim.x + threadIdx.x; idx < n;
       idx += gridDim.x * blockDim.x) {
    int t = idx / BINS, k = idx - t * BINS;
    size_t o = (size_t)t * NS + 2 * k;
    float er = expf(Ms[o]);
    float si, co;
    sincosf(Ms[o + 1], &si, &co);
    float mr = er * co, mi = er * si;
    float sr = S[o], sim = S[o + 1];
    S[o]     = sr * mr - sim * mi;
    S[o + 1] = sr * mi + sim * mr;
  }
}

// ---------------- overlap-add + window-square normalization ----------------
__global__ void k_final(const float* __restrict__ frames,
                        float* __restrict__ out) {
  for (int i = blockIdx.x * blockDim.x + threadIdx.x; i < OUTLEN;
       i += gridDim.x * blockDim.x) {
    int j = i + 512; // undo center pad
    int m0 = j >> 8, r = j & 255;
    float acc = 0.f, ws = 0.f;
    for (int d = 0; d < 4; ++d) {
      int m = m0 - d;
      if (m < 0 || m >= NT) continue;
      int n = r + 256 * d;
      acc += frames[(size_t)m * NFFT + n];
      float w = hannf(n);
      ws += w * w;
    }
    out[i] = acc / fmaxf(ws, 1e-11f);
  }
}

// ---------------- launcher ----------------
extern "C" void kernel_launch(void* const* d_in, const int* in_sizes, int n_in,
                              void* d_out, int out_size, void* d_ws,
                              size_t ws_size, hipStream_t stream) {
  const float* lf0     = (const float*)d_in[0]; // [1,16384]
  const float* env_noi = (const float*)d_in[1]; // [513,16384]
  const float* env_per = (const float*)d_in[2]; // [513,16384]
  const float* noise   = (const float*)d_in[3]; // [4194304]
  float* out = (float*)d_out;

  float* ws = (float*)d_ws;
  size_t off = 0;
  float* Bfwd  = ws + off; off += (size_t)KP_STFT * NS;
  float* B1    = ws + off; off += (size_t)KP_MP * NC;
  float* B2    = ws + off; off += (size_t)KP_MP * NS;
  float* Binv  = ws + off; off += (size_t)KP_IST * NFFT;
  float* f0up  = ws + off; off += (size_t)TS;
  float* saw   = ws + off; off += (size_t)TS;
  float* pulse = ws + off; off += (size_t)TS;
  float* bsums = ws + off; off += 1024;
  float* boffs = ws + off; off += 1024;
  float* Snoi  = ws + off; off += (size_t)NT * NS;
  float* Sper  = ws + off; off += (size_t)NT * NS;
  float* cepf  = ws + off; off += (size_t)NT * NC;
  float* Mspec = ws + off; off += (size_t)NT * NS;
  float* frames = ws + off; off += (size_t)NT * NFFT;
  if (ws_size < off * sizeof(float)) return;

  // 1. basis matrices (L2-resident, ~12 MB, pair-interleaved for b64 loads)
  k_build_bfwd<<<1024, 256, 0, stream>>>(Bfwd);
  k_build_b1<<<512, 256, 0, stream>>>(B1);
  k_build_b2<<<512, 256, 0, stream>>>(B2);
  k_build_binv<<<1024, 256, 0, stream>>>(Binv);

  // 2. impulse train
  k_f0a<<<1024, 256, 0, stream>>>(lf0, f0up, bsums);
  k_scan<<<1, 1024, 0, stream>>>(bsums, boffs);
  k_saw<<<1024, 256, 0, stream>>>(f0up, boffs, saw);
  k_pulse<<<2048, 256, 0, stream>>>(saw, f0up, pulse);

  // 3. STFTs (WMMA GEMM, reflect-padded frames, frame 0 dropped)
  k_gemm<2, NS><<<dim3(5, 1024), 256, 0, stream>>>(noise, Bfwd, Snoi, KP_STFT,
                                                   65, 0, 0, 0);
  k_gemm<2, NS><<<dim3(5, 1024), 256, 0, stream>>>(pulse, Bfwd, Sper, KP_STFT,
                                                   65, 0, 0, 0);

  // 4. noise envelope
  k_envnoi<<<2048, 256, 0, stream>>>(env_noi, Snoi);

  // 5. min-phase: cepstrum GEMM, re-FFT GEMM, complex exp * Sper
  k_gemm<1, NC><<<dim3(3, 1024), 256, 0, stream>>>(env_per, B1, cepf, KP_MP,
                                                   33, NT, BINS, 0);
  k_gemm<0, NS><<<dim3(5, 1024), 256, 0, stream>>>(cepf, B2, Mspec, KP_MP, 65,
                                                   NC, NC, 0);
  k_minph<<<2048, 256, 0, stream>>>(Mspec, Sper);

  // 6. iSTFT GEMMs into shared frame buffer (second pass accumulates)
  k_gemm<0, NFFT><<<dim3(4, 1024), 256, 0, stream>>>(Snoi, Binv, frames,
                                                     KP_IST, 64, NS, NS, 0);
  k_gemm<0, NFFT><<<dim3(4, 1024), 256, 0, stream>>>(Sper, Binv, frames,
                                                     KP_IST, 64, NS, NS, 1);

  // 7. overlap-add, /sum(w^2), trim pads
  k_final<<<2048, 256, 0, stream>>>(frames, out);
}